// RGCNEncoder_66735201845307
// MI455X (gfx1250) — compile-verified
//
#include <hip/hip_runtime.h>

// Problem constants (from the reference)
#define N_NODES 50000
#define R_REL   16
#define D_DIM   128
#define B_BLK   4
#define d_BLK   32
#define E_EDGES 800000

typedef __attribute__((ext_vector_type(2))) float v2f;
typedef __attribute__((ext_vector_type(8))) float v8f;

// ---------------------------------------------------------------- zero
__global__ void zero_f32(float* __restrict__ p, long n) {
    long i = (long)blockIdx.x * blockDim.x + threadIdx.x;
    if (i < n) p[i] = 0.0f;
}

// ------------------------------------------------------- per-(dst,rel) count
__global__ void count_kernel(const int* __restrict__ src, const int* __restrict__ dst,
                             const int* __restrict__ et, int* __restrict__ cnt) {
    int e = blockIdx.x * blockDim.x + threadIdx.x;
    if (e < E_EDGES) atomicAdd(&cnt[dst[e] * R_REL + et[e]], 1);
}

// ---------------------------------------------------- allR[r] = x @blockdiag W[r]
// One wave computes a 16(node)x16(col) f32 tile via 8 chained V_WMMA_F32_16X16X4_F32.
__global__ __launch_bounds__(256) void transform_wmma(const float* __restrict__ x,
                                                      const float* __restrict__ W,
                                                      float* __restrict__ allR) {
    int wave  = threadIdx.x >> 5;
    int lane  = threadIdx.x & 31;
    int combo = blockIdx.y * 8 + wave;    // 0..127 = (r, b, colHalf)
    int r  = combo >> 3;
    int b  = (combo >> 1) & 3;
    int ch = combo & 1;
    int m0 = blockIdx.x * 16;

    int half = lane >> 4;                 // K sub-pair selector
    int lid  = lane & 15;
    int rowA = m0 + lid;                  // A-matrix row (node)
    int colB = ch * 16 + lid;             // B/D column within the 32-wide block

    const float* Wrb  = W + (size_t)(r * B_BLK + b) * d_BLK * d_BLK; // [32 k][32 o]
    const float* xrow = x + (size_t)rowA * D_DIM + b * d_BLK;

    v8f c = {};
#pragma unroll
    for (int kc = 0; kc < 8; ++kc) {
        int k = kc * 4 + half * 2;
        v2f a;  a.x  = xrow[k];                 a.y  = xrow[k + 1];
        v2f bb; bb.x = Wrb[k * d_BLK + colB];   bb.y = Wrb[(k + 1) * d_BLK + colB];
        c = __builtin_amdgcn_wmma_f32_16x16x4_f32(false, a, false, bb,
                                                  (short)0, c, false, false);
    }

    float* out = allR + ((size_t)r * N_NODES + m0) * D_DIM + b * d_BLK + ch * 16 + lid;
#pragma unroll
    for (int vg = 0; vg < 8; ++vg) {
        int row = vg + half * 8;
        out[(size_t)row * D_DIM] = c[vg];
    }
}

// ------------------------------------------- edge gather + normalized scatter-add
// 32 lanes per edge, float4 per lane (128 cols), atomics land in L2-resident agg.
__global__ void scatter_kernel(const int* __restrict__ src, const int* __restrict__ dst,
                               const int* __restrict__ et, const int* __restrict__ cnt,
                               const float* __restrict__ allR, float* __restrict__ agg) {
    long t   = (long)blockIdx.x * blockDim.x + threadIdx.x;
    long e   = t >> 5;
    int lane = (int)(t & 31);
    if (e >= E_EDGES) return;
    int s = src[e], v = dst[e], r = et[e];
    float norm = 1.0f / fmaxf((float)cnt[v * R_REL + r], 1.0f);
    const float4* m = (const float4*)(allR + ((size_t)r * N_NODES + s) * D_DIM) + lane;
    float4 mv = *m;
    float* a = agg + (size_t)v * D_DIM + lane * 4;
    atomicAdd(a + 0, mv.x * norm);
    atomicAdd(a + 1, mv.y * norm);
    atomicAdd(a + 2, mv.z * norm);
    atomicAdd(a + 3, mv.w * norm);
}

// --------------------------- out = [relu](agg + x @ root + bias), root GEMM in WMMA
__global__ __launch_bounds__(256) void combine_wmma(const float* __restrict__ x,
                                                    const float* __restrict__ root,
                                                    const float* __restrict__ bias,
                                                    const float* __restrict__ agg,
                                                    float* __restrict__ out, int relu) {
    int wave = threadIdx.x >> 5;          // column tile 0..7
    int lane = threadIdx.x & 31;
    int m0   = blockIdx.x * 16;
    int half = lane >> 4;
    int lid  = lane & 15;
    int rowA = m0 + lid;
    int col  = wave * 16 + lid;

    const float* xrow = x + (size_t)rowA * D_DIM;

    v8f c = {};
#pragma unroll
    for (int kc = 0; kc < 32; ++kc) {
        int k = kc * 4 + half * 2;
        v2f a;  a.x  = xrow[k];               a.y  = xrow[k + 1];
        v2f bb; bb.x = root[k * D_DIM + col]; bb.y = root[(k + 1) * D_DIM + col];
        c = __builtin_amdgcn_wmma_f32_16x16x4_f32(false, a, false, bb,
                                                  (short)0, c, false, false);
    }

    float bv = bias[col];
#pragma unroll
    for (int vg = 0; vg < 8; ++vg) {
        int row = m0 + vg + half * 8;
        float v = c[vg] + agg[(size_t)row * D_DIM + col] + bv;
        if (relu) v = fmaxf(v, 0.0f);
        out[(size_t)row * D_DIM + col] = v;
    }
}

// ---------------------------------------------------------------- launcher
extern "C" void kernel_launch(void* const* d_in, const int* in_sizes, int n_in,
                              void* d_out, int out_size, void* d_ws, size_t ws_size,
                              hipStream_t stream) {
    const int*   edge_index = (const int*)d_in[0];
    const int*   src   = edge_index;
    const int*   dst   = edge_index + E_EDGES;
    const int*   et    = (const int*)d_in[1];
    const float* x0    = (const float*)d_in[2];
    const float* W1    = (const float*)d_in[3];
    const float* root1 = (const float*)d_in[4];
    const float* b1    = (const float*)d_in[5];
    const float* W2    = (const float*)d_in[6];
    const float* root2 = (const float*)d_in[7];
    const float* b2    = (const float*)d_in[8];

    // Workspace layout: cnt | agg | h | allR  (~464 MB total)
    char*  ws   = (char*)d_ws;
    int*   cnt  = (int*)ws;
    float* agg  = (float*)(ws + (size_t)N_NODES * R_REL * sizeof(int));
    float* h    = agg + (size_t)N_NODES * D_DIM;
    float* allR = h   + (size_t)N_NODES * D_DIM;

    const int ZB = 256;
    long zc = (long)N_NODES * R_REL + (long)N_NODES * D_DIM; // cnt+agg contiguous
    zero_f32<<<(unsigned)((zc + ZB - 1) / ZB), ZB, 0, stream>>>((float*)cnt, zc);
    count_kernel<<<(E_EDGES + ZB - 1) / ZB, ZB, 0, stream>>>(src, dst, et, cnt);

    dim3 tgrid(N_NODES / 16, (R_REL * B_BLK * 2) / 8);   // (3125, 16)
    dim3 sgrid((unsigned)(((long)E_EDGES * 32) / ZB));   // 100000 blocks

    // ---- Layer 1
    transform_wmma<<<tgrid, 256, 0, stream>>>(x0, W1, allR);
    scatter_kernel<<<sgrid, ZB, 0, stream>>>(src, dst, et, cnt, allR, agg);
    combine_wmma<<<N_NODES / 16, 256, 0, stream>>>(x0, root1, b1, agg, h, 1);

    // ---- Layer 2
    long za = (long)N_NODES * D_DIM;
    zero_f32<<<(unsigned)((za + ZB - 1) / ZB), ZB, 0, stream>>>(agg, za);
    transform_wmma<<<tgrid, 256, 0, stream>>>(h, W2, allR);
    scatter_kernel<<<sgrid, ZB, 0, stream>>>(src, dst, et, cnt, allR, agg);
    combine_wmma<<<N_NODES / 16, 256, 0, stream>>>(h, root2, b2, agg, (float*)d_out, 0);
}